// MultiViewGraphAttention_40785009443426
// MI455X (gfx1250) — compile-verified
//
#include <hip/hip_runtime.h>
#include <hip/hip_bf16.h>
#include <stdint.h>

#define N_NODES 50000
#define N_EDGES 800000
#define TOT_E   (N_EDGES + N_NODES)   // edges + self loops
#define HID     64
#define NHEAD   4
#define NEG_SLOPE 0.2f

typedef __attribute__((ext_vector_type(2))) float v2f;
typedef __attribute__((ext_vector_type(8))) float v8f;

// ---------------------------------------------------------------------------
// order-preserving float<->uint key for atomic segment-max (handles negatives)
// ---------------------------------------------------------------------------
__device__ __forceinline__ unsigned fkey(float f) {
  unsigned u = __float_as_uint(f);
  return (u & 0x80000000u) ? ~u : (u | 0x80000000u);
}
__device__ __forceinline__ float funkey(unsigned u) {
  return __uint_as_float((u & 0x80000000u) ? (u & 0x7FFFFFFFu) : ~u);
}

// ---------------------------------------------------------------------------
// fp32 WMMA GEMM: C[M x Ncols] = A[M x K] * B[K x Ncols] (+bias)
// grid = (M/16, Ncols/16), block = 32 (one wave -> one 16x16 tile)
// Uses V_WMMA_F32_16X16X4_F32: A 16x4 (2 VGPR), B 4x16 (2 VGPR), C 16x16 (8 VGPR)
//   A layout: lane<16 -> row=lane, k={k0,k0+1}; lane>=16 -> row=lane-16, k={k0+2,k0+3}
//   B layout: lane%16 -> col, lane/16 selects k-half (mirrors A)
//   C layout: vgpr v, lane<16 -> M=v, N=lane ; lane>=16 -> M=v+8, N=lane-16
// ---------------------------------------------------------------------------
__global__ __launch_bounds__(32)
void gemm16_f32(const float* __restrict__ A, const float* __restrict__ B,
                const float* __restrict__ bias, float* __restrict__ C,
                int K, int Ncols) {
  const int lane = threadIdx.x;
  const int half = lane >> 4;          // 0 or 1 (k-half)
  const int l    = lane & 15;
  const int row0 = blockIdx.x * 16;
  const int col0 = blockIdx.y * 16;

  const float* __restrict__ arow = A + (size_t)(row0 + l) * K + 2 * half;
  const float* __restrict__ bcol = B + (size_t)(2 * half) * Ncols + (col0 + l);

  v8f acc = {};
  for (int k0 = 0; k0 < K; k0 += 4) {
    v2f a = *(const v2f*)(arow + k0);                 // global_load_b64
    v2f b;
    b.x = bcol[(size_t)(k0    ) * Ncols];
    b.y = bcol[(size_t)(k0 + 1) * Ncols];
    acc = __builtin_amdgcn_wmma_f32_16x16x4_f32(
        /*neg_a=*/false, a, /*neg_b=*/false, b,
        /*c_mod=*/(short)0, acc, /*reuse_a=*/false, /*reuse_b=*/false);
  }

  const float bv = bias ? bias[col0 + l] : 0.0f;
#pragma unroll
  for (int v = 0; v < 8; ++v) {
    const int r = row0 + v + 8 * half;
    C[(size_t)r * Ncols + col0 + l] = acc[v] + bv;
  }
}

// ---------------------------------------------------------------------------
// per-(node, head) attention logits: e = <h[n,h,:], a[h,:]>
// ---------------------------------------------------------------------------
__global__ __launch_bounds__(256)
void attn_logits(const float* __restrict__ H, const float* __restrict__ a_src,
                 const float* __restrict__ a_dst,
                 float* __restrict__ e_src, float* __restrict__ e_dst) {
  const int p = blockIdx.x * blockDim.x + threadIdx.x;
  if (p >= N_NODES * NHEAD) return;
  const int node = p >> 2, head = p & 3;
  const float* __restrict__ hp = H + (size_t)node * (NHEAD * HID) + head * HID;
  const float* __restrict__ as = a_src + head * HID;
  const float* __restrict__ ad = a_dst + head * HID;
  float ss = 0.f, sd = 0.f;
#pragma unroll 8
  for (int c = 0; c < HID; ++c) { const float v = hp[c]; ss += v * as[c]; sd += v * ad[c]; }
  e_src[p] = ss;
  e_dst[p] = sd;
}

// ---------------------------------------------------------------------------
// generic u32 fill (zeros mkey/ssum/accum in one shot)
// ---------------------------------------------------------------------------
__global__ __launch_bounds__(256)
void fill_u32(unsigned* __restrict__ p, unsigned v, int n) {
  const int t = blockIdx.x * blockDim.x + threadIdx.x;
  if (t < n) p[t] = v;
}

__device__ __forceinline__ void edge_endpoints(const int* __restrict__ ei, int e,
                                               int& s, int& d) {
  if (e < N_EDGES) { s = ei[e]; d = ei[N_EDGES + e]; }  // edge_index is [2,E] row-major
  else             { s = e - N_EDGES; d = s; }          // self loop
}

// pass 1: segment max over destination (per head) via monotone-key atomicMax
__global__ __launch_bounds__(256)
void edge_max(const int* __restrict__ ei, const float* __restrict__ e_src,
              const float* __restrict__ e_dst, unsigned* __restrict__ mkey) {
  const int t = blockIdx.x * blockDim.x + threadIdx.x;
  if (t >= TOT_E * NHEAD) return;
  const int e = t >> 2, h = t & 3;
  int s, d; edge_endpoints(ei, e, s, d);
  float x = e_src[s * NHEAD + h] + e_dst[d * NHEAD + h];
  x = x > 0.f ? x : NEG_SLOPE * x;                      // leaky_relu
  atomicMax(&mkey[d * NHEAD + h], fkey(x));
}

// pass 2: w = exp(e - m[dst]); segment sum; cache w per edge
__global__ __launch_bounds__(256)
void edge_expsum(const int* __restrict__ ei, const float* __restrict__ e_src,
                 const float* __restrict__ e_dst, const unsigned* __restrict__ mkey,
                 float* __restrict__ ssum, float* __restrict__ wbuf) {
  const int t = blockIdx.x * blockDim.x + threadIdx.x;
  if (t >= TOT_E * NHEAD) return;
  const int e = t >> 2, h = t & 3;
  int s, d; edge_endpoints(ei, e, s, d);
  float x = e_src[s * NHEAD + h] + e_dst[d * NHEAD + h];
  x = x > 0.f ? x : NEG_SLOPE * x;
  const float m = funkey(mkey[d * NHEAD + h]);
  const float w = __expf(x - m);
  wbuf[t] = w;
  atomicAdd(&ssum[d * NHEAD + h], w);
}

// pass 3: one wave per edge; fold 1/H head-mean into alpha -> 64 atomics/edge
__global__ __launch_bounds__(256)
void edge_aggregate(const int* __restrict__ ei, const float* __restrict__ H,
                    const float* __restrict__ wbuf, const float* __restrict__ ssum,
                    float* __restrict__ accum) {
  const int warp = (blockIdx.x * blockDim.x + threadIdx.x) >> 5;   // wave32
  const int lane = threadIdx.x & 31;
  if (warp >= TOT_E) return;
  int s, d; edge_endpoints(ei, warp, s, d);
  const float* __restrict__ hs = H + (size_t)s * (NHEAD * HID);
  __builtin_prefetch(hs, 0, 1);                                    // global_prefetch_b8
  float al[NHEAD];
#pragma unroll
  for (int h = 0; h < NHEAD; ++h)
    al[h] = 0.25f * wbuf[warp * NHEAD + h] / (ssum[d * NHEAD + h] + 1e-16f);
#pragma unroll
  for (int cc = 0; cc < 2; ++cc) {
    const int c = lane + cc * 32;
    const float v = al[0] * hs[c] + al[1] * hs[HID + c] +
                    al[2] * hs[2 * HID + c] + al[3] * hs[3 * HID + c];
    atomicAdd(&accum[(size_t)d * HID + c], v);
  }
}

// ---------------------------------------------------------------------------
// finish: out = LayerNorm( ELU(accum + bias) + resid ) * g + be  (warp/node)
// ---------------------------------------------------------------------------
__global__ __launch_bounds__(256)
void finish_layer(const float* __restrict__ accum, const float* __restrict__ bias,
                  const float* __restrict__ resid, const float* __restrict__ g,
                  const float* __restrict__ be, float* __restrict__ out) {
  const int warp = (blockIdx.x * blockDim.x + threadIdx.x) >> 5;
  const int lane = threadIdx.x & 31;
  if (warp >= N_NODES) return;
  const size_t base = (size_t)warp * HID;

  float v0 = accum[base + lane]      + bias[lane];
  float v1 = accum[base + lane + 32] + bias[lane + 32];
  v0 = v0 > 0.f ? v0 : __expf(v0) - 1.f;   // ELU(alpha=1)
  v1 = v1 > 0.f ? v1 : __expf(v1) - 1.f;
  v0 += resid[base + lane];
  v1 += resid[base + lane + 32];

  float sum = v0 + v1;
#pragma unroll
  for (int o = 16; o > 0; o >>= 1) sum += __shfl_xor(sum, o, 32);
  const float mu = sum * (1.0f / HID);
  const float d0 = v0 - mu, d1 = v1 - mu;
  float vs = d0 * d0 + d1 * d1;
#pragma unroll
  for (int o = 16; o > 0; o >>= 1) vs += __shfl_xor(vs, o, 32);
  const float rstd = rsqrtf(vs * (1.0f / HID) + 1e-5f);

  out[base + lane]      = d0 * rstd * g[lane]      + be[lane];
  out[base + lane + 32] = d1 * rstd * g[lane + 32] + be[lane + 32];
}

// ---------------------------------------------------------------------------
// host side
// ---------------------------------------------------------------------------
extern "C" void kernel_launch(void* const* d_in, const int* in_sizes, int n_in,
                              void* d_out, int out_size, void* d_ws, size_t ws_size,
                              hipStream_t stream) {
  const float* x    = (const float*)d_in[0];
  const int*   ei   = (const int*)  d_in[1];
  const float* W_in = (const float*)d_in[2];
  const float* b_in = (const float*)d_in[3];
  const float* W1   = (const float*)d_in[4];
  const float* a1s  = (const float*)d_in[5];
  const float* a1d  = (const float*)d_in[6];
  const float* b1   = (const float*)d_in[7];
  const float* W2   = (const float*)d_in[8];
  const float* a2s  = (const float*)d_in[9];
  const float* a2d  = (const float*)d_in[10];
  const float* b2   = (const float*)d_in[11];
  const float* g1   = (const float*)d_in[12];
  const float* be1  = (const float*)d_in[13];
  const float* g2   = (const float*)d_in[14];
  const float* be2  = (const float*)d_in[15];
  const float* Wo   = (const float*)d_in[16];
  const float* bo   = (const float*)d_in[17];
  float* out = (float*)d_out;

  // workspace carve-up (float units, every offset even -> 8B aligned for v2f)
  float* ws = (float*)d_ws;
  float*    H     = ws;                                   // [N,256]
  float*    R     = H    + (size_t)N_NODES * 256;         // [N,64] residual / layer2 out
  float*    hcur  = R    + (size_t)N_NODES * 64;          // [N,64] layer1 out
  float*    esrc  = hcur + (size_t)N_NODES * 64;          // [N,4]
  float*    edst  = esrc + (size_t)N_NODES * 4;           // [N,4]
  unsigned* mkey  = (unsigned*)(edst + (size_t)N_NODES * 4); // [N,4]
  float*    ssum  = (float*)mkey + (size_t)N_NODES * 4;   // [N,4]
  float*    accum = ssum + (size_t)N_NODES * 4;           // [N,64]
  float*    wbuf  = accum + (size_t)N_NODES * 64;         // [E+N,4]

  const int initCnt = N_NODES * 4 + N_NODES * 4 + N_NODES * 64; // mkey+ssum+accum
  const dim3 blk256(256);
  const int eh_blocks  = (TOT_E * NHEAD + 255) / 256;
  const int agg_blocks = (TOT_E * 32 + 255) / 256;        // wave per edge
  const int fin_blocks = (N_NODES * 32 + 255) / 256;      // wave per node
  const int al_blocks  = (N_NODES * NHEAD + 255) / 256;
  const int ini_blocks = (initCnt + 255) / 256;

  // ---- layer 1 ----
  gemm16_f32<<<dim3(N_NODES / 16, 256 / 16), 32, 0, stream>>>(x, W1,  nullptr, H, 128, 256);
  gemm16_f32<<<dim3(N_NODES / 16,  64 / 16), 32, 0, stream>>>(x, W_in, b_in,   R, 128,  64);
  attn_logits<<<al_blocks, blk256, 0, stream>>>(H, a1s, a1d, esrc, edst);
  fill_u32<<<ini_blocks, blk256, 0, stream>>>(mkey, 0u, initCnt);
  edge_max   <<<eh_blocks, blk256, 0, stream>>>(ei, esrc, edst, mkey);
  edge_expsum<<<eh_blocks, blk256, 0, stream>>>(ei, esrc, edst, mkey, ssum, wbuf);
  edge_aggregate<<<agg_blocks, blk256, 0, stream>>>(ei, H, wbuf, ssum, accum);
  finish_layer<<<fin_blocks, blk256, 0, stream>>>(accum, b1, R, g1, be1, hcur);

  // ---- layer 2 ----
  gemm16_f32<<<dim3(N_NODES / 16, 256 / 16), 32, 0, stream>>>(hcur, W2, nullptr, H, 64, 256);
  attn_logits<<<al_blocks, blk256, 0, stream>>>(H, a2s, a2d, esrc, edst);
  fill_u32<<<ini_blocks, blk256, 0, stream>>>(mkey, 0u, initCnt);
  edge_max   <<<eh_blocks, blk256, 0, stream>>>(ei, esrc, edst, mkey);
  edge_expsum<<<eh_blocks, blk256, 0, stream>>>(ei, esrc, edst, mkey, ssum, wbuf);
  edge_aggregate<<<agg_blocks, blk256, 0, stream>>>(ei, H, wbuf, ssum, accum);
  finish_layer<<<fin_blocks, blk256, 0, stream>>>(accum, b2, hcur, g2, be2, R);

  // ---- output projection ----
  gemm16_f32<<<dim3(N_NODES / 16, 64 / 16), 32, 0, stream>>>(R, Wo, bo, out, 64, 64);
}